// FeatureFetcher_33861522162342
// MI455X (gfx1250) — compile-verified
//
#include <hip/hip_runtime.h>
#include <hip/hip_bf16.h>
#include <math.h>

// ---------------------------------------------------------------------------
// Problem constants (from reference)
// ---------------------------------------------------------------------------
#define HIDDEN 256
#define NHEADS 8
#define LVLS   4
#define NQ     300
#define BS     2
#define OUT    7
#define TOTAL  21760            // 128^2 + 64^2 + 32^2 + 16^2
#define BQ     (BS * NQ)        // 600
#define BQP    608              // 600 padded to multiple of 16
#define C4     (HIDDEN / 4)     // 64
#define P1K    (C4 * OUT * OUT) // 3136
#define MROIP  29408            // 600*49 padded to multiple of 16

typedef float v2f __attribute__((ext_vector_type(2)));
typedef float v8f __attribute__((ext_vector_type(8)));

// ---------------------------------------------------------------------------
// Branch-free WMMA f32 GEMM:  C = act(A[Mpad,K] * B[K,N] + bias[N])
// Compile-time N/K so all strides fold into immediate instruction offsets.
// blockDim.x = 32*WPB; each wave owns one 16x16 tile (same M tile, different
// N tile -> A lines fetched once per WGP).  Mpad, N multiples of 16; K%4==0.
// All outputs land in padded workspace -> no bounds checks; EXEC stays all-1s
// (required by WMMA).
// OUTMODE 0: C[row*N+col] ; OUTMODE 1: row=(bq,p) -> C[bq*3136 + col*49 + p]
// ACT: 0 none, 1 relu, 2 sigmoid, 3 tanh
// ---------------------------------------------------------------------------
template <int N, int K, int ACT, int OUTMODE>
__global__ __launch_bounds__(128)
void ff_gemm(const float* __restrict__ A, const float* __restrict__ B,
             const float* __restrict__ bias, float* __restrict__ C)
{
    const int tid   = threadIdx.x;
    const int wave  = tid >> 5;
    const int lane  = tid & 31;
    const int waves = blockDim.x >> 5;

    const int mbase = blockIdx.x * 16;
    const int nbase = (blockIdx.y * waves + wave) * 16;

    const int mh   = lane & 15;   // A-fragment row
    const int half = lane >> 4;   // 0 or 1 -> K pair select
    const int nl   = lane & 15;   // B/C-fragment col
    const int ka   = 2 * half;

    const float* __restrict__ Ar = A + (size_t)(mbase + mh) * K + ka;
    const float* __restrict__ Bc = B + (size_t)ka * N + (nbase + nl);

    v8f acc = {0.f, 0.f, 0.f, 0.f, 0.f, 0.f, 0.f, 0.f};

#pragma unroll 4
    for (int kk = 0; kk < K; kk += 4) {
        const v2f a = *reinterpret_cast<const v2f*>(Ar + kk);   // b64 load
        v2f b;
        b.x = Bc[(size_t)kk * N];           // compile-time stride -> IOFFSET
        b.y = Bc[(size_t)kk * N + N];
        acc = __builtin_amdgcn_wmma_f32_16x16x4_f32(
                  false, a, false, b, (short)0, acc, false, false);
    }

    const int col = nbase + nl;
    const float bv = bias[col];
#pragma unroll
    for (int v = 0; v < 8; ++v) {
        const int row = mbase + v + 8 * half;
        float x = acc[v] + bv;
        if      (ACT == 1) x = fmaxf(x, 0.0f);
        else if (ACT == 2) x = 1.0f / (1.0f + __expf(-x));
        else if (ACT == 3) x = tanhf(x);
        if (OUTMODE == 0) {
            C[(size_t)row * N + col] = x;
        } else {
            const int bq = row / 49;        // constant divisor -> mul/shift
            const int p  = row - bq * 49;
            C[(size_t)bq * P1K + (size_t)col * 49 + p] = x;
        }
    }
}

// ---------------------------------------------------------------------------
// Stage tgt into a padded (608-row) workspace buffer.
// ---------------------------------------------------------------------------
__global__ __launch_bounds__(256)
void ff_copy(const float* __restrict__ src, float* __restrict__ dst, int n)
{
    const int i = blockIdx.x * 256 + threadIdx.x;
    if (i < n) dst[i] = src[i];
}

// ---------------------------------------------------------------------------
// ROI-align (aligned=True, 7x7 out, sampling_ratio=2).
// One block per (b,q); 256 threads = 256 channels (channel-contiguous gathers
// since memory layout is (b, hw, c)).  Output layout: roi[(bq*49 + p)*256 + c]
// ---------------------------------------------------------------------------
__global__ __launch_bounds__(256)
void ff_roi_align(const float* __restrict__ memory, const float* __restrict__ ref,
                  float* __restrict__ roi, int H, int W, int start, int lvi)
{
    const int bq = blockIdx.x;            // 0..599
    const int b  = bq / NQ;
    const int t  = threadIdx.x;

    __shared__ float slx[14], sly[14], svx[14], svy[14];
    __shared__ int   sx0[14], sy0[14];

    if (t < 14) {
        const float* r = ref + ((size_t)bq * LVLS + lvi) * 6;
        const float cx = r[0], cy = r[1], le = r[2], to = r[3], ri = r[4], bo = r[5];
        const float x1 = fminf(fmaxf(cx - le, 0.f), 1.f) * (float)W;
        const float y1 = fminf(fmaxf(cy - to, 0.f), 1.f) * (float)H;
        const float x2 = fminf(fmaxf(cx + ri, 0.f), 1.f) * (float)W;
        const float y2 = fminf(fmaxf(cy + bo, 0.f), 1.f) * (float)H;
        const float sx = x1 - 0.5f, sy = y1 - 0.5f;
        const float bw = (x2 - x1) * (1.0f / OUT);
        const float bh = (y2 - y1) * (1.0f / OUT);
        const float tv = (float)(t >> 1) + ((float)(t & 1) + 0.5f) * 0.5f;
        const float xx = sx + bw * tv;
        const float yy = sy + bh * tv;
        svx[t] = (xx >= -1.0f && xx <= (float)W) ? 1.0f : 0.0f;
        svy[t] = (yy >= -1.0f && yy <= (float)H) ? 1.0f : 0.0f;
        const float xc = fminf(fmaxf(xx, 0.f), (float)W - 1.0f);
        const float yc = fminf(fmaxf(yy, 0.f), (float)H - 1.0f);
        const float x0 = fminf(fmaxf(floorf(xc), 0.f), (float)W - 2.0f);
        const float y0 = fminf(fmaxf(floorf(yc), 0.f), (float)H - 2.0f);
        slx[t] = xc - x0;  sly[t] = yc - y0;
        sx0[t] = (int)x0;  sy0[t] = (int)y0;
    }
    __syncthreads();

    const int c = t;  // channel
    const float* mb = memory + ((size_t)b * TOTAL + start) * HIDDEN + c;
    const size_t rowStride = (size_t)W * HIDDEN;

    for (int oy = 0; oy < OUT; ++oy) {
        for (int ox = 0; ox < OUT; ++ox) {
            float acc = 0.0f;
#pragma unroll
            for (int s2 = 0; s2 < 4; ++s2) {
                const int rI = oy * 2 + (s2 >> 1);
                const int cI = ox * 2 + (s2 & 1);
                const float vmask = svy[rI] * svx[cI];   // uniform per block
                if (vmask > 0.0f) {
                    const int   y0 = sy0[rI], x0 = sx0[cI];
                    const float ly = sly[rI], lx = slx[cI];
                    const float* p00 = mb + (size_t)(y0 * W + x0) * HIDDEN;
                    const float m00 = p00[0];
                    const float m01 = p00[HIDDEN];
                    const float m10 = p00[rowStride];
                    const float m11 = p00[rowStride + HIDDEN];
                    acc += (1.f - ly) * (1.f - lx) * m00 + (1.f - ly) * lx * m01
                         + ly * (1.f - lx) * m10 + ly * lx * m11;
                }
            }
            roi[((size_t)bq * 49 + oy * OUT + ox) * HIDDEN + c] = acc * 0.25f;
        }
    }
}

// ---------------------------------------------------------------------------
// grid_sample (bilinear, zeros, align_corners=False) on the 7x7 ROI at 8 head
// points, multiplied by the per-head gate.  qe[(bq*32 + lvi*8 + h)*256 + c]
// ---------------------------------------------------------------------------
__global__ __launch_bounds__(256)
void ff_grid_sample(const float* __restrict__ roi, const float* __restrict__ pts,
                    const float* __restrict__ gate, float* __restrict__ qe, int lvi)
{
    const int bq = blockIdx.x;
    const int c  = threadIdx.x;
#pragma unroll
    for (int h = 0; h < NHEADS; ++h) {
        const float gx = pts[bq * 16 + h * 2 + 0];
        const float gy = pts[bq * 16 + h * 2 + 1];
        const float ix = ((gx + 1.0f) * (float)OUT - 1.0f) * 0.5f;
        const float iy = ((gy + 1.0f) * (float)OUT - 1.0f) * 0.5f;
        const float x0f = floorf(ix), y0f = floorf(iy);
        const float lx = ix - x0f, ly = iy - y0f;
        const int x0 = (int)x0f, y0 = (int)y0f;
        float acc = 0.0f;
#pragma unroll
        for (int dy = 0; dy < 2; ++dy) {
#pragma unroll
            for (int dx = 0; dx < 2; ++dx) {
                const int xi = x0 + dx, yi = y0 + dy;
                if (xi >= 0 && xi < OUT && yi >= 0 && yi < OUT) {
                    const float w = (dy ? ly : 1.0f - ly) * (dx ? lx : 1.0f - lx);
                    acc += w * roi[((size_t)bq * 49 + yi * OUT + xi) * HIDDEN + c];
                }
            }
        }
        const float g = gate[((size_t)bq * NHEADS + h) * HIDDEN + c];
        qe[((size_t)bq * (LVLS * NHEADS) + lvi * NHEADS + h) * HIDDEN + c] = acc * g;
    }
}

// ---------------------------------------------------------------------------
// Final: logits[32] = qe . attn2_w + b ; softmax ; out = sum_j qe[j]*wgt[j]
// 8 waves x 4 logits via wave32 shuffle reductions; single barrier.
// ---------------------------------------------------------------------------
__global__ __launch_bounds__(256)
void ff_final(const float* __restrict__ qe, const float* __restrict__ w2,
              const float* __restrict__ b2, float* __restrict__ out)
{
    const int bq   = blockIdx.x;
    const int t    = threadIdx.x;
    const int wave = t >> 5;
    const int lane = t & 31;
    __shared__ float lg[LVLS * NHEADS];

    const float* __restrict__ base = qe + (size_t)bq * (LVLS * NHEADS) * HIDDEN;

#pragma unroll
    for (int jj = 0; jj < 4; ++jj) {
        const int j = wave * 4 + jj;
        const float* row = base + (size_t)j * HIDDEN;
        float s = 0.0f;
#pragma unroll
        for (int c = 0; c < HIDDEN; c += 32) s += row[c + lane] * w2[c + lane];
#pragma unroll
        for (int off = 16; off > 0; off >>= 1) s += __shfl_xor(s, off, 32);
        if (lane == 0) lg[j] = s + b2[0];
    }
    __syncthreads();

    // softmax weights computed redundantly per thread (32 entries, broadcast reads)
    float mx = lg[0];
#pragma unroll
    for (int j = 1; j < LVLS * NHEADS; ++j) mx = fmaxf(mx, lg[j]);
    float sum = 0.0f;
#pragma unroll
    for (int j = 0; j < LVLS * NHEADS; ++j) sum += __expf(lg[j] - mx);
    const float inv = 1.0f / sum;

    float acc = 0.0f;
#pragma unroll
    for (int j = 0; j < LVLS * NHEADS; ++j)
        acc += base[(size_t)j * HIDDEN + t] * (__expf(lg[j] - mx) * inv);
    out[(size_t)bq * HIDDEN + t] = acc;
}

// ---------------------------------------------------------------------------
// Host launcher
// ---------------------------------------------------------------------------
extern "C" void kernel_launch(void* const* d_in, const int* in_sizes, int n_in,
                              void* d_out, int out_size, void* d_ws, size_t ws_size,
                              hipStream_t stream)
{
    (void)in_sizes; (void)n_in; (void)out_size; (void)ws_size;

    const float* tgt     = (const float*)d_in[0];
    const float* memory  = (const float*)d_in[1];
    const float* ref     = (const float*)d_in[2];
    /* d_in[3] = level_start_index (compile-time constants used) */
    const float* p1_w    = (const float*)d_in[4];
    const float* p1_b    = (const float*)d_in[5];
    const float* p2_w1   = (const float*)d_in[6];
    const float* p2_b1   = (const float*)d_in[7];
    const float* p2_w2   = (const float*)d_in[8];
    const float* p2_b2   = (const float*)d_in[9];
    const float* p2_w3   = (const float*)d_in[10];
    const float* p2_b3   = (const float*)d_in[11];
    const float* p2_w4   = (const float*)d_in[12];
    const float* p2_b4   = (const float*)d_in[13];
    const float* attn1_w = (const float*)d_in[14];
    const float* attn1_b = (const float*)d_in[15];
    const float* attn2_w = (const float*)d_in[16];
    const float* attn2_b = (const float*)d_in[17];
    float* out = (float*)d_out;

    // Workspace layout (floats) -- all GEMM operands padded to 16-row multiples
    float* ws   = (float*)d_ws;
    float* tgtp = ws;                              // 608*256
    float* gate = tgtp + (size_t)BQP * HIDDEN;     // 608*2048
    float* roi  = gate + (size_t)BQP * 2048;       // 29408*256
    float* p0   = roi  + (size_t)MROIP * HIDDEN;   // 608*3136
    float* pA   = p0   + (size_t)BQP * P1K;        // 608*256
    float* pB   = pA   + (size_t)BQP * 256;        // 608*512
    float* pC   = pB   + (size_t)BQP * 512;        // 608*512
    float* pts  = pC   + (size_t)BQP * 512;        // 608*16
    float* qe   = pts  + (size_t)BQP * 16;         // 600*32*256

    const int lvlH[LVLS]   = {128, 64, 32, 16};
    const int lvlW[LVLS]   = {128, 64, 32, 16};
    const int starts[LVLS] = {0, 16384, 20480, 21504};

    // 0) stage tgt into padded buffer
    ff_copy<<<(BQ * HIDDEN + 255) / 256, 256, 0, stream>>>(tgt, tgtp, BQ * HIDDEN);

    // 1) gate = sigmoid(tgt @ attn1_w + attn1_b)   (608 x 2048, K=256)
    ff_gemm<2048, HIDDEN, 2, 0><<<dim3(BQP / 16, 2048 / 64), 128, 0, stream>>>(
        tgtp, attn1_w, attn1_b, gate);

    for (int lv = 0; lv < LVLS; ++lv) {
        // 2a) ROI align -> roi (bq, p, c)
        ff_roi_align<<<BQ, 256, 0, stream>>>(memory, ref, roi,
                                             lvlH[lv], lvlW[lv], starts[lv], lv);
        // 2b) channel projection: (29408 x 64, K=256), relu, scatter d*49+p
        ff_gemm<C4, HIDDEN, 1, 1><<<dim3(MROIP / 16, 1), 128, 0, stream>>>(
            roi, p1_w, p1_b, p0);
        // 2c) MLP chain
        ff_gemm<256, P1K, 1, 0><<<dim3(BQP / 16, 256 / 64), 128, 0, stream>>>(
            p0, p2_w1, p2_b1, pA);
        ff_gemm<512, 256, 1, 0><<<dim3(BQP / 16, 512 / 64), 128, 0, stream>>>(
            pA, p2_w2, p2_b2, pB);
        ff_gemm<512, 512, 1, 0><<<dim3(BQP / 16, 512 / 64), 128, 0, stream>>>(
            pB, p2_w3, p2_b3, pC);
        ff_gemm<16, 512, 3, 0><<<dim3(BQP / 16, 1), 32, 0, stream>>>(
            pC, p2_w4, p2_b4, pts);
        // 2d) grid-sample + gate -> qe
        ff_grid_sample<<<BQ, 256, 0, stream>>>(roi, pts, gate, qe, lv);
    }

    // 3) softmax-weighted sum over the 32 (level,head) entries
    ff_final<<<BQ, 256, 0, stream>>>(qe, attn2_w, attn2_b, out);
}